// Group_987842478294
// MI455X (gfx1250) — compile-verified
//
#include <hip/hip_runtime.h>
#include <hip/hip_bf16.h>

#define B_ 32
#define N_ 8192
#define G_ 512
#define K_ 32

typedef float v2f __attribute__((ext_vector_type(2)));
typedef float v8f __attribute__((ext_vector_type(8)));
typedef int v4i __attribute__((ext_vector_type(4)));

// Async global->LDS staging (CDNA5 ASYNCcnt path), guarded so a name mismatch
// degrades to a synchronous LDS copy instead of a compile failure.
#if defined(__has_builtin)
#if __has_builtin(__builtin_amdgcn_global_load_async_to_lds_b128) && \
    __has_builtin(__builtin_amdgcn_s_wait_asynccnt)
#define USE_ASYNC_LDS 1
// Builtin signature (from hipcc diagnostic): param0 is int4* in the global
// address space; param1 is the LDS-side pointer.
typedef __attribute__((address_space(1))) v4i gas_v4i;
typedef __attribute__((address_space(3))) v4i las_v4i;
#endif
#endif

// ---------------------------------------------------------------------------
// Kernel 1: furthest point sampling. One block per batch, 1024 threads.
// All 8192 points + running min-dists live in registers (8 per thread).
// Argmax via packed (dist_bits<<32 | ~idx) max-reduction: wave32 shuffles,
// then one ds-atomic max into a double-buffered LDS slot (2 barriers/step).
// ---------------------------------------------------------------------------
#define FPS_T 1024
#define PT 8  // points per thread

__global__ __launch_bounds__(FPS_T) void fps_kernel(const float* __restrict__ xyz,
                                                    float* __restrict__ centerOut) {
    const int b = blockIdx.x;
    const int tid = threadIdx.x;
    const float* base = xyz + (size_t)b * N_ * 3;
    float* cOut = centerOut + (size_t)b * G_ * 3;

    float px[PT], py[PT], pz[PT], dd[PT];
#pragma unroll
    for (int i = 0; i < PT; ++i) {
        int p = tid * PT + i;
        px[i] = base[p * 3 + 0];
        py[i] = base[p * 3 + 1];
        pz[i] = base[p * 3 + 2];
        dd[i] = 1e10f;
    }

    __shared__ float sCx, sCy, sCz;
    __shared__ unsigned long long sMax[2];

    if (tid == 0) {  // pointnet2 semantics: first center is point 0
        sCx = px[0]; sCy = py[0]; sCz = pz[0];
        cOut[0] = px[0]; cOut[1] = py[0]; cOut[2] = pz[0];
        sMax[0] = 0ull; sMax[1] = 0ull;
    }
    __syncthreads();

    const int lane = tid & 31;

    for (int g = 1; g < G_; ++g) {
        const int s = g & 1;
        const float cx = sCx, cy = sCy, cz = sCz;
        unsigned long long best = 0ull;
#pragma unroll
        for (int i = 0; i < PT; ++i) {
            float dx = px[i] - cx, dy = py[i] - cy, dz = pz[i] - cz;
            float d = dx * dx + dy * dy + dz * dz;
            d = fminf(dd[i], d);
            dd[i] = d;
            // squared dist >= 0 -> float bits are order-preserving.
            // ~idx in low bits: max-reduce picks SMALLEST index on ties
            // (matches jnp.argmax first-index semantics).
            unsigned long long pk =
                ((unsigned long long)__float_as_uint(d) << 32) |
                (unsigned int)~(unsigned int)(tid * PT + i);
            best = (pk > best) ? pk : best;
        }
#pragma unroll
        for (int off = 16; off > 0; off >>= 1) {
            unsigned long long o = __shfl_xor(best, off, 32);
            best = (o > best) ? o : best;
        }
        if (lane == 0) atomicMax(&sMax[s], best);
        // Reset the slot the NEXT iteration will use. Safe: the previous
        // iteration's readers of sMax[s^1] are separated by its end barrier.
        if (tid == 0) sMax[s ^ 1] = 0ull;
        __syncthreads();
        const int idx = (int)~(unsigned int)(sMax[s] & 0xffffffffull);
        if ((idx >> 3) == tid) {  // owner thread broadcasts the new center
            const int i = idx & (PT - 1);
            sCx = px[i]; sCy = py[i]; sCz = pz[i];
            cOut[g * 3 + 0] = px[i];
            cOut[g * 3 + 1] = py[i];
            cOut[g * 3 + 2] = pz[i];
        }
        __syncthreads();
    }
}

// ---------------------------------------------------------------------------
// Kernel 2: WMMA distance tiles + threshold-gated top-32 + gather.
// One block (256 threads = 8 waves) per (batch, 16-center tile).
// d'[g,n] = |p_n|^2 - 2 c_g . p_n  (dropping |c|^2 preserves top-k order):
//   A[m,:] = (-2cx, -2cy, -2cz, 1)   B[:,n] = (px, py, pz, |p|^2)
// via V_WMMA_F32_16X16X4_F32 -> 16x16 distance tile per wave.
// Point chunks are staged into LDS double-buffered: async DMA for chunk i+1
// overlaps WMMA + selection of chunk i.
// ---------------------------------------------------------------------------
#define GT 16     // centers per block
#define NCH 128   // points per chunk per block (8 waves * 16)
#define NCPY (NCH * 3 / 4)  // 96 float4 copies per chunk

__global__ __launch_bounds__(256) void knn_group_kernel(
    const float* __restrict__ xyz, const float* __restrict__ color,
    const float* __restrict__ centerAll, float* __restrict__ outNeigh,
    float* __restrict__ outFeat) {
    const int b = blockIdx.x >> 5;   // / (G_/GT)
    const int gt = blockIdx.x & 31;  // tile index within batch
    const int tid = threadIdx.x;
    const int lane = tid & 31;
    const int wave = tid >> 5;

    __shared__ float sPts[2][NCH * 3];
    __shared__ float sCen[GT * 3];
    __shared__ float sD[GT * NCH];
    __shared__ float sCandD[GT * NCH];
    __shared__ int sCandI[GT * NCH];
    __shared__ int sCandCnt[GT];
    __shared__ float sTopD[GT * K_];
    __shared__ int sTopI[GT * K_];
    __shared__ int sCnt[GT];
    __shared__ float sKth[GT];

    const float* pbase = xyz + (size_t)b * N_ * 3;

    const float* cen = centerAll + ((size_t)b * G_ + (size_t)gt * GT) * 3;
    if (tid < GT * 3) sCen[tid] = cen[tid];
    if (tid < GT) {
        sCandCnt[tid] = 0;
        sCnt[tid] = 0;
        sKth[tid] = 3.4e38f;
    }

    // Stage a 128-point chunk (96 x float4 = 1.5KB) into LDS buffer `buf`.
    auto stage = [&](int buf, int cb) {
        if (tid < NCPY) {
            const float* gsrc = pbase + (size_t)cb * 3 + (size_t)tid * 4;
            float* ldst = &sPts[buf][tid * 4];
#if defined(USE_ASYNC_LDS)
            __builtin_amdgcn_global_load_async_to_lds_b128(
                (gas_v4i*)(void*)gsrc, (las_v4i*)(void*)ldst, 0, 0);
#else
            const float4 v = *(const float4*)gsrc;
            *(float4*)ldst = v;
#endif
        }
    };

    stage(0, 0);
#if defined(USE_ASYNC_LDS)
    __builtin_amdgcn_s_wait_asynccnt(0);
#endif
    __syncthreads();

    // A fragment: 16x4 f32, lanes 0-15 carry (K0,K1), lanes 16-31 carry (K2,K3)
    v2f afrag;
    if (lane < 16) {
        afrag.x = -2.0f * sCen[lane * 3 + 0];
        afrag.y = -2.0f * sCen[lane * 3 + 1];
    } else {
        afrag.x = -2.0f * sCen[(lane - 16) * 3 + 2];
        afrag.y = 1.0f;
    }

    int cur = 0;
    for (int cb = 0; cb < N_; cb += NCH) {
        const int pl = wave * 16 + (lane & 15);  // chunk-local point index
        const float qx = sPts[cur][pl * 3 + 0];
        const float qy = sPts[cur][pl * 3 + 1];
        const float qz = sPts[cur][pl * 3 + 2];
        const float q2 = qx * qx + qy * qy + qz * qz;
        v2f bfrag;
        bfrag.x = (lane < 16) ? qx : qz;
        bfrag.y = (lane < 16) ? qy : q2;

        // Kick off the DMA for the next chunk; it overlaps WMMA + selection.
        if (cb + NCH < N_) stage(cur ^ 1, cb + NCH);

        v8f acc = {};
        acc = __builtin_amdgcn_wmma_f32_16x16x4_f32(
            /*neg_a=*/false, afrag, /*neg_b=*/false, bfrag,
            /*c_mod=*/(short)0, acc, /*reuse_a=*/false, /*reuse_b=*/false);

        // Scatter 16x16 tile: VGPR r -> center r (lanes 0-15) / r+8 (lanes 16-31)
        const int col = pl;
        const int rbase = (lane < 16) ? 0 : 8;
#pragma unroll
        for (int r = 0; r < 8; ++r) sD[(rbase + r) * NCH + col] = acc[r];
        __syncthreads();

        // Candidate scan: team c = 16 threads per center, threshold = current
        // 32nd-smallest (inf until the list is full -> never miss a candidate).
        const int c = tid >> 4, m = tid & 15;
        const float kth = sKth[c];
#pragma unroll
        for (int j = m; j < NCH; j += 16) {
            const float d = sD[c * NCH + j];
            if (d < kth) {
                const int pos = atomicAdd(&sCandCnt[c], 1);
                sCandD[c * NCH + pos] = d;
                sCandI[c * NCH + pos] = cb + j;
            }
        }
        __syncthreads();

        // One thread per center: sorted insertion of survivors (ascending).
        if (m == 0) {
            int cnt = sCnt[c];
            const int nc = sCandCnt[c];
            for (int i2 = 0; i2 < nc; ++i2) {
                const float d = sCandD[c * NCH + i2];
                const int id = sCandI[c * NCH + i2];
                if (cnt < K_) {
                    int pos = cnt;
                    while (pos > 0 && sTopD[c * K_ + pos - 1] > d) {
                        sTopD[c * K_ + pos] = sTopD[c * K_ + pos - 1];
                        sTopI[c * K_ + pos] = sTopI[c * K_ + pos - 1];
                        --pos;
                    }
                    sTopD[c * K_ + pos] = d;
                    sTopI[c * K_ + pos] = id;
                    ++cnt;
                } else if (d < sTopD[c * K_ + K_ - 1]) {
                    int pos = K_ - 1;
                    while (pos > 0 && sTopD[c * K_ + pos - 1] > d) {
                        sTopD[c * K_ + pos] = sTopD[c * K_ + pos - 1];
                        sTopI[c * K_ + pos] = sTopI[c * K_ + pos - 1];
                        --pos;
                    }
                    sTopD[c * K_ + pos] = d;
                    sTopI[c * K_ + pos] = id;
                }
            }
            sCnt[c] = cnt;
            sKth[c] = (cnt >= K_) ? sTopD[c * K_ + K_ - 1] : 3.4e38f;
            sCandCnt[c] = 0;
        }
#if defined(USE_ASYNC_LDS)
        __builtin_amdgcn_s_wait_asynccnt(0);
#endif
        __syncthreads();  // selection done + next chunk's LDS data visible
        cur ^= 1;
    }

    // Gather + write neighborhood / features.
    const float* clbase = color + (size_t)b * N_ * 3;
    for (int p = tid; p < GT * K_; p += 256) {
        const int c = p >> 5, k = p & 31;
        const int g = gt * GT + c;
        const int idx = sTopI[c * K_ + k];
        const float x = pbase[idx * 3 + 0] - sCen[c * 3 + 0];
        const float y = pbase[idx * 3 + 1] - sCen[c * 3 + 1];
        const float z = pbase[idx * 3 + 2] - sCen[c * 3 + 2];
        const size_t on = (((size_t)b * G_ + g) * K_ + k) * 3;
        outNeigh[on + 0] = x;
        outNeigh[on + 1] = y;
        outNeigh[on + 2] = z;
        const size_t of = (((size_t)b * G_ + g) * K_ + k) * 6;
        outFeat[of + 0] = x;
        outFeat[of + 1] = y;
        outFeat[of + 2] = z;
        outFeat[of + 3] = clbase[idx * 3 + 0];
        outFeat[of + 4] = clbase[idx * 3 + 1];
        outFeat[of + 5] = clbase[idx * 3 + 2];
    }
}

extern "C" void kernel_launch(void* const* d_in, const int* in_sizes, int n_in,
                              void* d_out, int out_size, void* d_ws, size_t ws_size,
                              hipStream_t stream) {
    (void)in_sizes; (void)n_in; (void)out_size; (void)d_ws; (void)ws_size;
    const float* xyz = (const float*)d_in[0];
    const float* color = (const float*)d_in[1];
    float* out = (float*)d_out;
    float* outNeigh = out;                                    // [B,G,K,3]
    float* outCenter = out + (size_t)B_ * G_ * K_ * 3;        // [B,G,3]
    float* outFeat = outCenter + (size_t)B_ * G_ * 3;         // [B,G,K,6]

    fps_kernel<<<B_, FPS_T, 0, stream>>>(xyz, outCenter);
    knn_group_kernel<<<B_ * (G_ / GT), 256, 0, stream>>>(xyz, color, outCenter,
                                                         outNeigh, outFeat);
}